// TopkPPR_48112223650276
// MI455X (gfx1250) — compile-verified
//
#include <hip/hip_runtime.h>
#include <stdint.h>
#include <stddef.h>

#define N_NODES 50000
#define N_EDGES 800000
#define NSRC    256
#define TOPK    64
#define ALPHA   0.15f
#define NITER   30
#define TILE    16      // dst nodes per workgroup == WMMA M
#define CHUNK_E 512     // CSR edges staged into LDS per round (multiple of 4)

typedef float v2f __attribute__((ext_vector_type(2)));
typedef float v8f __attribute__((ext_vector_type(8)));

// CDNA5 async global->LDS staging (ASYNCcnt path), with safe fallback.
#if defined(__HIP_DEVICE_COMPILE__) && \
    __has_builtin(__builtin_amdgcn_global_load_async_to_lds_b32) && \
    __has_builtin(__builtin_amdgcn_s_wait_asynccnt)
#define USE_ASYNC_LDS 1
typedef __attribute__((address_space(1))) int g_int;   // global (device) int
typedef __attribute__((address_space(3))) int l_int;   // LDS int
#else
#define USE_ASYNC_LDS 0
#endif

// ---------------- preprocessing ----------------

__global__ __launch_bounds__(256) void zero3_kernel(float* deg, int* incnt, int* fill, int n) {
  int i = blockIdx.x * blockDim.x + threadIdx.x;
  if (i < n) { deg[i] = 0.f; incnt[i] = 0; fill[i] = 0; }
}

__global__ __launch_bounds__(256) void deg_kernel(const long long* __restrict__ ei,
                                                  const float* __restrict__ ew,
                                                  float* __restrict__ deg, int e) {
  int i = blockIdx.x * blockDim.x + threadIdx.x;
  if (i < e) atomicAdd(&deg[(int)ei[i]], ew[i]);
}

__global__ __launch_bounds__(256) void wnorm_kernel(const long long* __restrict__ ei,
                                                    const float* __restrict__ ew,
                                                    const float* __restrict__ deg,
                                                    float* __restrict__ wn,
                                                    int* __restrict__ incnt, int e) {
  int i = blockIdx.x * blockDim.x + threadIdx.x;
  if (i < e) {
    int s = (int)ei[i];
    int d = (int)ei[N_EDGES + i];
    float dg = fmaxf(deg[s], 1e-12f);
    wn[i] = ew[i] / dg;
    atomicAdd(&incnt[d], 1);
  }
}

// single-workgroup chunked exclusive scan (runs once; n = 50000)
__global__ __launch_bounds__(256) void scan_kernel(const int* __restrict__ in,
                                                   int* __restrict__ out, int n) {
  __shared__ int buf[256];
  __shared__ int carry;
  int tid = threadIdx.x;
  if (tid == 0) carry = 0;
  __syncthreads();
  for (int base = 0; base < n; base += 256) {
    int x = (base + tid < n) ? in[base + tid] : 0;
    buf[tid] = x;
    __syncthreads();
    for (int offt = 1; offt < 256; offt <<= 1) {
      int t = (tid >= offt) ? buf[tid - offt] : 0;
      __syncthreads();
      buf[tid] += t;
      __syncthreads();
    }
    if (base + tid < n) out[base + tid] = carry + buf[tid] - x;
    __syncthreads();
    if (tid == 0) carry += buf[255];
    __syncthreads();
  }
  if (tid == 0) out[n] = carry;
}

__global__ __launch_bounds__(256) void scatter_kernel(const long long* __restrict__ ei,
                                                      const float* __restrict__ wn,
                                                      const int* __restrict__ rp,
                                                      int* __restrict__ fill,
                                                      int* __restrict__ col,
                                                      float* __restrict__ val,
                                                      int* __restrict__ dl, int e) {
  int i = blockIdx.x * blockDim.x + threadIdx.x;
  if (i < e) {
    int s = (int)ei[i];
    int d = (int)ei[N_EDGES + i];
    int pos = atomicAdd(&fill[d], 1);
    int j = rp[d] + pos;
    col[j] = s;
    val[j] = wn[i];
    dl[j]  = d & (TILE - 1);
  }
}

// p0: node-major [N][NSRC]; idx[s] == s so p0[n][s] = (n==s) ? ALPHA : 0
__global__ __launch_bounds__(256) void init_p_kernel(float* __restrict__ p) {
  size_t i = (size_t)blockIdx.x * 256 + threadIdx.x;
  if (i < (size_t)N_NODES * NSRC) {
    int n = (int)(i / NSRC);
    int s = (int)(i % NSRC);
    p[i] = (n == s) ? ALPHA : 0.f;
  }
}

// ---------------- WMMA SpMM iteration ----------------
// p_new[d][s] = p0[d][s] + (1-ALPHA) * sum_{e: dst==d} w[e] * p_old[src[e]][s]
// One workgroup per 16-dst tile; wave w owns source chunks 2w, 2w+1 (16 cols each).
// Per 4-edge group: A[16x4] = edge weight placed in its local-dst row (1 nz/col),
// B[4x16] = gathered p_old rows. C accumulates the 16x16 dst-by-source tile.
// Edge arrays in LDS are padded to a multiple of 4 so the inner loop is guard-free
// (wave-uniform control flow; EXEC all-ones around the WMMAs, per ISA 7.12).
__global__ __launch_bounds__(256)
void spmm_wmma_kernel(const float* __restrict__ p_old, float* __restrict__ p_new,
                      const int* __restrict__ rp, const int* __restrict__ col,
                      const float* __restrict__ val, const int* __restrict__ dl) {
  __shared__ int   s_src[CHUNK_E];
  __shared__ float s_w[CHUNK_E];
  __shared__ int   s_dl[CHUNK_E];

  const int tile = blockIdx.x;
  const int nodeBase = tile * TILE;
  const int tid  = threadIdx.x;
  const int lane = tid & 31;
  const int wave = tid >> 5;        // 0..7
  const int cb   = wave * 32;       // first source column of this wave's 2 chunks

  // A fragment slots (16x4 f32): VGPR0 lanes0-15 K=0 / lanes16-31 K=2; VGPR1 K=1 / K=3
  const int m   = lane & 15;
  const int kA0 = (lane < 16) ? 0 : 2;
  const int kA1 = (lane < 16) ? 1 : 3;
  // B fragment slots (4x16 f32): VGPR0 rows K=0/1 across half-waves; VGPR1 rows K=2/3
  const int kB  = lane >> 4;
  const int nb  = lane & 15;

  v8f acc0 = {}; v8f acc1 = {};

  const int e0 = rp[nodeBase];
  const int e1 = rp[nodeBase + TILE];

  for (int base = e0; base < e1; base += CHUNK_E) {
    int n = e1 - base; if (n > CHUNK_E) n = CHUNK_E;
    int nPad = (n + 3) & ~3;

#if USE_ASYNC_LDS
    for (int i = tid; i < nPad; i += 256) {
      if (i < n) {
        __builtin_amdgcn_global_load_async_to_lds_b32(
            (g_int*)(col + base + i), (l_int*)&s_src[i], 0, 0);
        __builtin_amdgcn_global_load_async_to_lds_b32(
            (g_int*)(val + base + i), (l_int*)&s_w[i], 0, 0);
        __builtin_amdgcn_global_load_async_to_lds_b32(
            (g_int*)(dl + base + i), (l_int*)&s_dl[i], 0, 0);
      } else {
        s_src[i] = 0; s_w[i] = 0.f; s_dl[i] = -1;
      }
    }
    __builtin_amdgcn_s_wait_asynccnt(0);   // drain this wave's ASYNCcnt before the barrier
#else
    for (int i = tid; i < nPad; i += 256) {
      bool ok = i < n;
      s_src[i] = ok ? col[base + i] : 0;
      s_w[i]   = ok ? val[base + i] : 0.f;
      s_dl[i]  = ok ? dl[base + i]  : -1;
    }
#endif
    __syncthreads();

    int ngE = nPad >> 2;              // uniform across the workgroup
    for (int g = 0; g < ngE; ++g) {
      int eb = g << 2;
      // guard-free LDS reads (pad slots carry w=0 / dl=-1 / src=0)
      float wA0 = s_w[eb + kA0];
      float wA1 = s_w[eb + kA1];
      int   dA0 = s_dl[eb + kA0];
      int   dA1 = s_dl[eb + kA1];
      v2f a;
      a.x = (dA0 == m) ? wA0 : 0.f;
      a.y = (dA1 == m) ? wA1 : 0.f;

      unsigned r0 = (unsigned)s_src[eb + kB]     * NSRC + cb + nb;
      unsigned r1 = (unsigned)s_src[eb + kB + 2] * NSRC + cb + nb;
      v2f b0, b1;
      b0.x = p_old[r0];      b0.y = p_old[r1];
      b1.x = p_old[r0 + 16]; b1.y = p_old[r1 + 16];

      acc0 = __builtin_amdgcn_wmma_f32_16x16x4_f32(false, a, false, b0,
                                                   (short)0, acc0, false, false);
      acc1 = __builtin_amdgcn_wmma_f32_16x16x4_f32(false, a, false, b1,
                                                   (short)0, acc1, false, false);
    }
    __syncthreads();
  }

  // C/D layout: VGPR r -> lanes0-15 (M=r, N=lane), lanes16-31 (M=r+8, N=lane-16)
  int Nc   = lane & 15;
  int mAdd = (lane < 16) ? 0 : 8;
#pragma unroll
  for (int r = 0; r < 8; ++r) {
    int node = nodeBase + r + mAdd;
    int s0 = cb + Nc;
    int s1 = s0 + 16;
    float v0 = ((node == s0) ? ALPHA : 0.f) + (1.f - ALPHA) * acc0[r];
    float v1 = ((node == s1) ? ALPHA : 0.f) + (1.f - ALPHA) * acc1[r];
    p_new[(size_t)node * NSRC + s0] = v0;
    p_new[(size_t)node * NSRC + s1] = v1;
  }
}

// ---------------- transpose [N][NSRC] -> [NSRC][N] for contiguous top-k scans ----------------
__global__ __launch_bounds__(1024)
void transpose_kernel(const float* __restrict__ p, float* __restrict__ pT) {
  __shared__ float t[32][33];
  int nBase = blockIdx.x * 32;
  int sBase = blockIdx.y * 32;
  int x = threadIdx.x, y = threadIdx.y;
  int n = nBase + y;
  t[y][x] = (n < N_NODES) ? p[(size_t)n * NSRC + sBase + x] : 0.f;
  __syncthreads();
  int no = nBase + x;
  if (no < N_NODES) pT[(size_t)(sBase + y) * N_NODES + no] = t[x][y];
}

// ---------------- per-source top-64: radix-select threshold + bitonic sort ----------------
__global__ __launch_bounds__(256)
void topk_kernel(const float* __restrict__ pT, const float* __restrict__ deg,
                 float* __restrict__ out_w, float* __restrict__ out_c) {
  const int s   = blockIdx.x;
  const int tid = threadIdx.x;
  const float* row = pT + (size_t)s * N_NODES;

  __shared__ int      red[256];
  __shared__ unsigned scnt;
  __shared__ float    cval[256];
  __shared__ int      cidx[256];

  // all values >= 0, so float order == uint bit order: binary-search the K-th largest pattern
  unsigned X = 0u;
  for (int b = 30; b >= 0; --b) {
    unsigned t = X | (1u << b);
    int cnt = 0;
    for (int n = tid; n < N_NODES; n += 256)
      cnt += (__float_as_uint(row[n]) >= t) ? 1 : 0;
    red[tid] = cnt;
    __syncthreads();
    for (int off = 128; off > 0; off >>= 1) {
      if (tid < off) red[tid] += red[tid + off];
      __syncthreads();
    }
    if (red[0] >= TOPK) X = t;
    __syncthreads();
  }

  // gather candidates (value >= X); count >= TOPK by construction
  if (tid == 0) scnt = 0;
  __syncthreads();
  for (int n = tid; n < N_NODES; n += 256) {
    float v = row[n];
    if (__float_as_uint(v) >= X) {
      unsigned slot = atomicAdd(&scnt, 1u);
      if (slot < 256u) { cval[slot] = v; cidx[slot] = n; }
    }
  }
  __syncthreads();
  unsigned m = scnt; if (m > 256u) m = 256u;
  if ((unsigned)tid >= m) { cval[tid] = -1.f; cidx[tid] = 0x7FFFFFFF; }
  __syncthreads();

  // bitonic sort of 256 entries: value desc, index asc (matches top_k tie-break)
  for (int k = 2; k <= 256; k <<= 1) {
    for (int j = k >> 1; j > 0; j >>= 1) {
      int ixj = tid ^ j;
      if (ixj > tid) {
        bool up = ((tid & k) == 0);
        float va = cval[tid], vb = cval[ixj];
        int   ia = cidx[tid], ib = cidx[ixj];
        bool b_before_a = (vb > va) || (vb == va && ib < ia);
        if (b_before_a == up) {
          cval[tid] = vb; cval[ixj] = va;
          cidx[tid] = ib; cidx[ixj] = ia;
        }
      }
      __syncthreads();
    }
  }

  if (tid < TOPK) {
    float dsrc = sqrtf(fmaxf(deg[s], 1e-12f));   // idx[s] == s
    int ci = cidx[tid];
    int cc = (ci == 0x7FFFFFFF) ? 0 : ci;
    float dinv = rsqrtf(fmaxf(deg[cc], 1e-12f));
    float v = cval[tid]; if (v < 0.f) v = 0.f;
    out_w[s * TOPK + tid] = dsrc * v * dinv;
    out_c[s * TOPK + tid] = (float)cc;
  }
}

// ---------------- driver ----------------

extern "C" void kernel_launch(void* const* d_in, const int* in_sizes, int n_in,
                              void* d_out, int out_size, void* d_ws, size_t ws_size,
                              hipStream_t stream) {
  (void)in_sizes; (void)n_in; (void)out_size; (void)ws_size;
  const long long* edge_index  = (const long long*)d_in[0];  // [2, E] int64
  const float*     edge_weight = (const float*)d_in[1];      // [E] f32
  // d_in[2] is idx = arange(NSRC); identity, folded into the kernels

  char* ws = (char*)d_ws;
  size_t off = 0;
  auto carve = [&](size_t bytes) -> void* {
    void* p = ws + off;
    off += (bytes + 255) & ~(size_t)255;
    return p;
  };
  float* deg    = (float*)carve((size_t)N_NODES * 4);
  int*   incnt  = (int*)  carve((size_t)N_NODES * 4);
  int*   fill   = (int*)  carve((size_t)N_NODES * 4);
  int*   rowptr = (int*)  carve((size_t)(N_NODES + 1) * 4);
  float* wnorm  = (float*)carve((size_t)N_EDGES * 4);
  int*   colA   = (int*)  carve((size_t)N_EDGES * 4);
  float* valA   = (float*)carve((size_t)N_EDGES * 4);
  int*   dlA    = (int*)  carve((size_t)N_EDGES * 4);
  float* pA     = (float*)carve((size_t)N_NODES * NSRC * 4);
  float* pB     = (float*)carve((size_t)N_NODES * NSRC * 4);

  const int NB = (N_NODES + 255) / 256;
  const int EB = (N_EDGES + 255) / 256;

  zero3_kernel  <<<NB, 256, 0, stream>>>(deg, incnt, fill, N_NODES);
  deg_kernel    <<<EB, 256, 0, stream>>>(edge_index, edge_weight, deg, N_EDGES);
  wnorm_kernel  <<<EB, 256, 0, stream>>>(edge_index, edge_weight, deg, wnorm, incnt, N_EDGES);
  scan_kernel   <<<1, 256, 0, stream>>>(incnt, rowptr, N_NODES);
  scatter_kernel<<<EB, 256, 0, stream>>>(edge_index, wnorm, rowptr, fill, colA, valA, dlA, N_EDGES);
  init_p_kernel <<<N_NODES, 256, 0, stream>>>(pA);

  float* pin = pA;
  float* pout = pB;
  for (int it = 0; it < NITER; ++it) {
    spmm_wmma_kernel<<<N_NODES / TILE, 256, 0, stream>>>(pin, pout, rowptr, colA, valA, dlA);
    float* t = pin; pin = pout; pout = t;
  }
  // NITER even -> final state in pA (pin); reuse pout as the transposed buffer
  float* pT = pout;
  dim3 tb(32, 32);
  dim3 tg((N_NODES + 31) / 32, NSRC / 32);
  transpose_kernel<<<tg, tb, 0, stream>>>(pin, pT);

  float* out_w = (float*)d_out;                    // [NSRC, K] weights
  float* out_c = (float*)d_out + NSRC * TOPK;      // [NSRC, K] neighbor ids (as float)
  topk_kernel<<<NSRC, 256, 0, stream>>>(pT, deg, out_w, out_c);
}